// VAELoss_31439160606725
// MI455X (gfx1250) — compile-verified
//
#include <hip/hip_runtime.h>
#include <hip/hip_bf16.h>

typedef __attribute__((ext_vector_type(2))) float v2f;
typedef __attribute__((ext_vector_type(8))) float v8f;

#define B_SZ   1024
#define S_SZ   277
#define R_SZ   76
#define L_SZ   24
#define Z_SZ   56
#define NTOK   (B_SZ * S_SZ)      // 283648 tokens
#define NTILE  (NTOK / 16)        // 17728 tiles of 16 tokens
#define NCHUNK (R_SZ / 4)         // 19 K-chunks of 4 for WMMA f32 16x16x4
// torch clamps natural logs at -100; we work in log2 domain and rescale by ln2.
#define LOG2_CLAMP (-144.26950408889634f)   // -100 / ln(2)
#define LN2        (0.6931471805599453f)

// ---------------------------------------------------------------------------
// BCE kernel: fused argmax -> mask gather -> WMMA row-sum -> renorm -> logs.
// One wave handles a 16-token tile; recon_x row tile lives in registers in
// WMMA-A layout (lane<16: K=4c,4c+1 ; lane>=16: K=4c+2,4c+3 for token m=lane&15).
// ---------------------------------------------------------------------------
__global__ __launch_bounds__(256) void vae_bce_kernel(
    const float* __restrict__ x,
    const float* __restrict__ recon,
    const float* __restrict__ masks,
    const int*   __restrict__ i2l,
    float*       __restrict__ acc_out)
{
    __shared__ float m_lds[L_SZ * R_SZ];
    __shared__ int   i2l_lds[R_SZ];
    __shared__ float red[8];

    const int tid = threadIdx.x;
    for (int i = tid; i < L_SZ * R_SZ; i += 256) m_lds[i] = masks[i];
    if (tid < R_SZ) i2l_lds[tid] = i2l[tid];
    __syncthreads();

    const int  lane  = tid & 31;
    const int  wib   = tid >> 5;                  // wave in block (8 waves)
    const int  gwave = blockIdx.x * 8 + wib;
    const int  nwav  = gridDim.x * 8;

    const int  m    = lane & 15;                  // token within tile
    const bool hi   = lane >= 16;                 // upper half-lane group
    const int  ksub = hi ? 2 : 0;

    float acc = 0.0f;   // accumulates base-2 log terms

    for (int tile = gwave; tile < NTILE; tile += nwav) {
        const int token = tile * 16 + m;
        const float* __restrict__ xrow = x     + (size_t)token * R_SZ;
        const float* __restrict__ rrow = recon + (size_t)token * R_SZ;

        // Load recon tile (kept in registers) and scan x for the one-hot index.
        v2f xm[NCHUNK];
        int ti_local = -1;
#pragma unroll
        for (int c = 0; c < NCHUNK; ++c) {
            const int k0 = 4 * c + ksub;
            const float xv0 = xrow[k0];
            const float xv1 = xrow[k0 + 1];
            v2f rv;
            rv.x = rrow[k0];
            rv.y = rrow[k0 + 1];
            xm[c] = rv;
            if (xv0 > 0.5f) ti_local = k0;
            if (xv1 > 0.5f) ti_local = k0 + 1;
        }
        // Each half-lane pair covers disjoint K; combine to get the token's index.
        const int ti = max(ti_local, __shfl_xor(ti_local, 16, 32));
        const int lhs = i2l_lds[ti];
        const float* __restrict__ mrow = &m_lds[lhs * R_SZ];

        // Apply mask in place (tile stays in WMMA-A layout).
#pragma unroll
        for (int c = 0; c < NCHUNK; ++c) {
            const int k0 = 4 * c + ksub;
            xm[c].x *= mrow[k0];
            xm[c].y *= mrow[k0 + 1];
        }

        // Row sums: D += A(16x4 f32) x Ones(4x16) over 19 chunks.
        v2f ones;
        ones.x = 1.0f;
        ones.y = 1.0f;
        v8f d = {};
#pragma unroll
        for (int c = 0; c < NCHUNK; ++c) {
            d = __builtin_amdgcn_wmma_f32_16x16x4_f32(
                    false, xm[c], false, ones, (short)0, d, false, false);
        }

        // Extract this token's sum from the C/D layout:
        //   VGPR j holds M=j (lanes 0-15) or M=j+8 (lanes 16-31).
        const int j = m & 7;
        float mine = d[0];
#pragma unroll
        for (int jj = 1; jj < 8; ++jj) mine = (j == jj) ? d[jj] : mine;
        const float other = __shfl_xor(mine, 16, 32);
        const bool need_partner = ((m >= 8) != hi);
        const float rowsum = need_partner ? other : mine;
        const float inv = __builtin_amdgcn_rcpf(rowsum);  // v_rcp_f32

        // BCE terms (log2 domain): log2(1-o) everywhere; patch in log2(o) at
        // the true index. v_log_f32 co-executes with the VALU stream.
#pragma unroll
        for (int c = 0; c < NCHUNK; ++c) {
            const int k0 = 4 * c + ksub;
            const float o0 = xm[c].x * inv;
            const float o1 = xm[c].y * inv;
            const float l20 = fmaxf(__log2f(1.0f - o0), LOG2_CLAMP);
            const float l21 = fmaxf(__log2f(1.0f - o1), LOG2_CLAMP);
            acc += l20 + l21;
            if (k0 == ti)     acc += fmaxf(__log2f(o0), LOG2_CLAMP) - l20;
            if (k0 + 1 == ti) acc += fmaxf(__log2f(o1), LOG2_CLAMP) - l21;
        }
    }

    acc *= LN2;  // back to natural-log domain

    // Wave reduction, then block reduction, one atomic per block.
#pragma unroll
    for (int off = 16; off > 0; off >>= 1) acc += __shfl_xor(acc, off, 32);
    if (lane == 0) red[wib] = acc;
    __syncthreads();
    if (tid == 0) {
        float s = 0.0f;
#pragma unroll
        for (int w = 0; w < 8; ++w) s += red[w];
        atomicAdd(acc_out, -s);   // bce = -sum(log terms)
    }
}

// ---------------------------------------------------------------------------
// KLD kernel: -0.5 * sum(1 + log_var - mu^2 - exp(log_var)) over B*Z elems.
// ---------------------------------------------------------------------------
__global__ __launch_bounds__(256) void vae_kld_kernel(
    const float* __restrict__ mu,
    const float* __restrict__ lv,
    float*       __restrict__ acc_out,
    int n)
{
    __shared__ float red[8];
    const int tid = threadIdx.x;
    const int i = blockIdx.x * 256 + tid;
    float t = 0.0f;
    if (i < n) {
        const float mv = mu[i];
        const float lvv = lv[i];
        t = 1.0f + lvv - mv * mv - __expf(lvv);
    }
#pragma unroll
    for (int off = 16; off > 0; off >>= 1) t += __shfl_xor(t, off, 32);
    if ((tid & 31) == 0) red[tid >> 5] = t;
    __syncthreads();
    if (tid == 0) {
        float s = 0.0f;
#pragma unroll
        for (int w = 0; w < 8; ++w) s += red[w];
        atomicAdd(acc_out, -0.5f * s);
    }
}

__global__ void vae_zero_kernel(float* ws) { ws[0] = 0.0f; }

__global__ void vae_final_kernel(const float* __restrict__ ws,
                                 float* __restrict__ out)
{
    out[0] = ws[0] * (1.0f / (float)B_SZ);
}

// ---------------------------------------------------------------------------
extern "C" void kernel_launch(void* const* d_in, const int* in_sizes, int n_in,
                              void* d_out, int out_size, void* d_ws, size_t ws_size,
                              hipStream_t stream) {
    (void)in_sizes; (void)n_in; (void)out_size; (void)ws_size;
    const float* x     = (const float*)d_in[0];
    const float* mu    = (const float*)d_in[1];
    const float* lv    = (const float*)d_in[2];
    const float* recon = (const float*)d_in[3];
    const float* masks = (const float*)d_in[4];
    const int*   i2l   = (const int*)d_in[5];
    float* out = (float*)d_out;
    float* ws  = (float*)d_ws;

    vae_zero_kernel<<<1, 1, 0, stream>>>(ws);

    // 17728 tiles; 1108 blocks * 8 waves = 8864 waves -> exactly 2 tiles/wave.
    vae_bce_kernel<<<dim3(1108), dim3(256), 0, stream>>>(x, recon, masks, i2l, ws);

    const int nkld = B_SZ * Z_SZ;  // 57344 -> 224 blocks
    vae_kld_kernel<<<dim3((nkld + 255) / 256), dim3(256), 0, stream>>>(mu, lv, ws, nkld);

    vae_final_kernel<<<1, 1, 0, stream>>>(ws, out);
}